// LSTM_70308614635605
// MI455X (gfx1250) — compile-verified
//
#include <hip/hip_runtime.h>
#include <hip/hip_bf16.h>

#ifndef __has_builtin
#define __has_builtin(x) 0
#endif

// Problem constants (from reference): x is [B, T, D] f32, UNITS = 1.
#define B_TOT  4096
#define T_TOT  2048
#define D_IN   8
#define CT     16                    // timesteps per LDS chunk
#define NCHUNK (T_TOT / CT)          // 128
#define SEG_F  (CT * D_IN)           // 128 floats (512 B) per sequence per chunk
#define PITCH  (SEG_F + 4)           // 132 dwords: 16B-aligned, bank-skewed
#define NLANE  32                    // wave32: one lane per batch element

typedef int v4i __attribute__((ext_vector_type(4)));
typedef __attribute__((address_space(1))) v4i glb_v4i_t;
typedef __attribute__((address_space(3))) v4i lds_v4i_t;

// Issue NSEG async b128 copies with compile-time global byte offsets so the
// 64KB segment stride lands in the instruction's imm24 IOFFSET field instead
// of per-segment v_add_nc_u64 address math.
template <int S>
struct Issuer {
    static __device__ __forceinline__ void run(const float* g, float* l) {
        Issuer<S - 1>::run(g, l);
#if __has_builtin(__builtin_amdgcn_global_load_async_to_lds_b128)
        __builtin_amdgcn_global_load_async_to_lds_b128(
            (glb_v4i_t*)(unsigned long long)g,
            (lds_v4i_t*)(unsigned int)(unsigned long long)(l + (S - 1) * PITCH),
            /*imm byte offset*/ (S - 1) * T_TOT * D_IN * 4,
            /*cpol*/ 0);
#else
        *(float4*)(l + (S - 1) * PITCH) =
            *(const float4*)(g + (size_t)(S - 1) * T_TOT * D_IN);
#endif
    }
};
template <>
struct Issuer<0> {
    static __device__ __forceinline__ void run(const float*, float*) {}
};

__device__ __forceinline__ float fast_sigmoid(float x) {
    return 1.0f / (1.0f + __expf(-x));
}

__device__ __forceinline__ float fast_tanh(float x) {
    float a = fabsf(x);
    float e = __expf(-2.0f * a);          // e in (0,1], no overflow
    float t = (1.0f - e) / (1.0f + e);
    return copysignf(t, x);
}

__global__ __launch_bounds__(NLANE)
void lstm_scan_kernel(const float* __restrict__ x,
                      const float* __restrict__ kernel,     // [8,4]
                      const float* __restrict__ rec_kernel, // [1,4]
                      const float* __restrict__ bias,       // [4]
                      const float* __restrict__ W1,         // [1,64]
                      const float* __restrict__ b1,         // [64]
                      const float* __restrict__ W2,         // [64,5]
                      const float* __restrict__ b2,         // [5]
                      float* __restrict__ out)              // [B,5]
{
    __shared__ float tile[2][NLANE * PITCH];                // 2 x 16.5 KB

    const int lane = threadIdx.x;                           // 0..31
    const int b0   = blockIdx.x * NLANE;
    const int b    = b0 + lane;

    // Small weights -> registers (uniform addresses -> scalar loads).
    float kr[D_IN][4];
    #pragma unroll
    for (int d = 0; d < D_IN; ++d)
        #pragma unroll
        for (int g = 0; g < 4; ++g)
            kr[d][g] = kernel[d * 4 + g];
    float rk[4], bs[4];
    #pragma unroll
    for (int g = 0; g < 4; ++g) { rk[g] = rec_kernel[g]; bs[g] = bias[g]; }

    // Running per-lane global base: lane's 16B quad within segment 0 of the
    // current chunk; advances by 512 B (one chunk) per iteration.
    const float* gbase = x + (size_t)b0 * T_TOT * D_IN + (size_t)lane * 4;

    // ---- prefetch chunk 0: 32 async b128 ops, each a full 512B segment ----
    Issuer<NLANE>::run(gbase, &tile[0][lane * 4]);
    gbase += SEG_F;

    float h = 0.0f, c = 0.0f;

    for (int k = 0; k < NCHUNK; ++k) {
        const int buf = k & 1;

        if (k + 1 < NCHUNK) {
            // Issue next chunk into the other buffer, then wait until only
            // those 32 remain outstanding (async loads retire in order).
            Issuer<NLANE>::run(gbase, &tile[buf ^ 1][lane * 4]);
            gbase += SEG_F;
            asm volatile("s_wait_asynccnt 0x20" ::: "memory");
        } else {
            asm volatile("s_wait_asynccnt 0x0" ::: "memory");
        }

        // ---- recurrence over CT steps from LDS (bank-conflict-free b128) ----
        const float* seg = &tile[buf][lane * PITCH];
        #pragma unroll
        for (int t = 0; t < CT; ++t) {
            float4 q0 = *(const float4*)(seg + t * D_IN);
            float4 q1 = *(const float4*)(seg + t * D_IN + 4);
            float xd[D_IN] = {q0.x, q0.y, q0.z, q0.w, q1.x, q1.y, q1.z, q1.w};

            float z[4];
            #pragma unroll
            for (int g = 0; g < 4; ++g) z[g] = fmaf(h, rk[g], bs[g]);
            #pragma unroll
            for (int d = 0; d < D_IN; ++d)
                #pragma unroll
                for (int g = 0; g < 4; ++g)
                    z[g] = fmaf(xd[d], kr[d][g], z[g]);

            float ig = fast_sigmoid(z[0]);
            float fg = fast_sigmoid(z[1]);
            float gg = fast_tanh(z[2]);
            float og = fast_sigmoid(z[3]);
            c = fmaf(fg, c, ig * gg);
            h = og * fast_tanh(c);
        }
    }

    // ---- Dense(64, relu) -> Dense(5) -> softmax (scalar h, per thread) ----
    float acc[5];
    #pragma unroll
    for (int j = 0; j < 5; ++j) acc[j] = b2[j];
    #pragma unroll 8
    for (int j = 0; j < 64; ++j) {
        float hid = fmaf(h, W1[j], b1[j]);
        hid = fmaxf(hid, 0.0f);
        #pragma unroll
        for (int kk = 0; kk < 5; ++kk)
            acc[kk] = fmaf(hid, W2[j * 5 + kk], acc[kk]);
    }
    float m = acc[0];
    #pragma unroll
    for (int kk = 1; kk < 5; ++kk) m = fmaxf(m, acc[kk]);
    float sum = 0.0f;
    #pragma unroll
    for (int kk = 0; kk < 5; ++kk) { acc[kk] = __expf(acc[kk] - m); sum += acc[kk]; }
    float inv = 1.0f / sum;
    #pragma unroll
    for (int kk = 0; kk < 5; ++kk) out[(size_t)b * 5 + kk] = acc[kk] * inv;
}

extern "C" void kernel_launch(void* const* d_in, const int* in_sizes, int n_in,
                              void* d_out, int out_size, void* d_ws, size_t ws_size,
                              hipStream_t stream) {
    (void)in_sizes; (void)n_in; (void)out_size; (void)d_ws; (void)ws_size;
    const float* x   = (const float*)d_in[0];
    const float* ker = (const float*)d_in[1];
    const float* rec = (const float*)d_in[2];
    const float* bia = (const float*)d_in[3];
    const float* W1  = (const float*)d_in[4];
    const float* b1  = (const float*)d_in[5];
    const float* W2  = (const float*)d_in[6];
    const float* b2  = (const float*)d_in[7];
    float* out = (float*)d_out;

    dim3 grid(B_TOT / NLANE);   // 128 single-wave workgroups, one per sequence-group
    dim3 block(NLANE);          // wave32
    hipLaunchKernelGGL(lstm_scan_kernel, grid, block, 0, stream,
                       x, ker, rec, bia, W1, b1, W2, b2, out);
}